// BatchSimilarityMetrics_69286412419139
// MI455X (gfx1250) — compile-verified
//
#include <hip/hip_runtime.h>
#include <hip/hip_bf16.h>

typedef float v2f __attribute__((ext_vector_type(2)));
typedef float v8f __attribute__((ext_vector_type(8)));

#define NPB (3 * 256 * 256)   // elements per batch item
#define NBINS 256

// ---------------------------------------------------------------------------
// block-wide f32 sum reduction (wave32 shuffles + LDS partials)
// ---------------------------------------------------------------------------
__device__ inline float block_reduce_sum(float v, float* sred) {
  __syncthreads();
  #pragma unroll
  for (int off = 16; off; off >>= 1) v += __shfl_xor(v, off, 32);
  const int wid  = threadIdx.x >> 5;
  const int lane = threadIdx.x & 31;
  if (lane == 0) sred[wid] = v;
  __syncthreads();
  const int nw = (blockDim.x + 31) >> 5;
  if (wid == 0) {
    float r = (lane < nw) ? sred[lane] : 0.0f;
    #pragma unroll
    for (int off = 16; off; off >>= 1) r += __shfl_xor(r, off, 32);
    if (lane == 0) sred[0] = r;
  }
  __syncthreads();
  float res = sred[0];
  __syncthreads();
  return res;
}

// ---------------------------------------------------------------------------
// Kernel 1: fused mutual information (256x256 joint histogram in the 320 KB
// WGP LDS) + PSNR squared-error accumulation. One workgroup per batch item.
// ---------------------------------------------------------------------------
__global__ void mi_psnr_kernel(const float* __restrict__ x,
                               const float* __restrict__ y,
                               float* __restrict__ out) {
  extern __shared__ unsigned int hist[];   // 65536 bins = 256 KB dynamic LDS
  __shared__ float sred[32];

  const int b   = blockIdx.x;
  const int tid = threadIdx.x;             // blockDim.x == 1024
  const float invN = 1.0f / (float)NPB;

  for (int i = tid; i < NBINS * NBINS; i += 1024) hist[i] = 0u;
  __syncthreads();

  const float* xb = x + (size_t)b * NPB;
  const float* yb = y + (size_t)b * NPB;

  float se = 0.0f;                         // PSNR squared error partial
  for (int i = tid; i < NPB; i += 1024) {
    const float xv = xb[i];
    const float yv = yb[i];
    const float d = (xv - yv) * 0.5f;      // (x*0.5+0.5)-(y*0.5+0.5)
    se += d * d;
    int ix = (int)((xv + 1.0f) * 0.5f * 256.0f);
    int iy = (int)((yv + 1.0f) * 0.5f * 256.0f);
    ix = ix < 0 ? 0 : (ix > 255 ? 255 : ix);
    iy = iy < 0 ? 0 : (iy > 255 ? 255 : iy);
    atomicAdd(&hist[ix * NBINS + iy], 1u); // ds_add_u32
  }
  __syncthreads();

  // joint entropy partial
  float hxy_p = 0.0f;
  for (int i = tid; i < NBINS * NBINS; i += 1024) {
    const unsigned c = hist[i];
    if (c) { const float p = (float)c * invN; hxy_p -= p * log2f(p); }
  }
  // marginal entropies: thread i (<256) owns row i (px) and column i (py)
  float hx_p = 0.0f, hy_p = 0.0f;
  if (tid < NBINS) {
    unsigned rc = 0, cc = 0;
    for (int j = 0; j < NBINS; ++j) {
      rc += hist[tid * NBINS + j];
      cc += hist[j * NBINS + tid];
    }
    if (rc) { const float p = (float)rc * invN; hx_p -= p * log2f(p); }
    if (cc) { const float p = (float)cc * invN; hy_p -= p * log2f(p); }
  }

  const float hxy = block_reduce_sum(hxy_p, sred);
  const float hx  = block_reduce_sum(hx_p,  sred);
  const float hy  = block_reduce_sum(hy_p,  sred);
  const float sse = block_reduce_sum(se,    sred);

  if (tid == 0) {
    // MI = Hx + Hy - Hxy  (== sum jp*log2(jp/(px*py)) on the support of jp)
    float mi = hx + hy - hxy;
    const float norm = fminf(hx, hy);
    mi = (norm > 0.0f) ? mi / norm : 0.0f;
    mi = fminf(fmaxf(mi, 0.0f), 1.0f);
    out[b * 3 + 0] = mi;

    const float mse = sse * invN;
    const float psnr = (mse == 0.0f) ? 100.0f : (-10.0f * log10f(mse));
    out[b * 3 + 2] = psnr / 40.0f;
  }
}

// ---------------------------------------------------------------------------
// Kernel 2: SSIM. One workgroup (256 threads = 8 waves) per (b,c) image.
// Separable 11-tap Gaussian: horizontal pass into 5 LDS bands (28 rows x 256
// cols), vertical pass as banded 16x16 matmul on V_WMMA_F32_16X16X4_F32
// (7 chained K=4 chunks, exact f32).
// ---------------------------------------------------------------------------
__global__ void ssim_kernel(const float* __restrict__ x,
                            const float* __restrict__ y,
                            const float* __restrict__ win,
                            float* __restrict__ ws_partial) {
  extern __shared__ float ldsf[];          // 5 * 28 * 256 floats = 140 KB
  __shared__ float sred[32];
  __shared__ float gsh[11];                // 1-D gaussian taps

  const int img = blockIdx.x;              // b*3 + c
  const int tid = threadIdx.x;             // 256 threads
  const size_t base = (size_t)img * 65536; // 256*256
  const float* xb = x + base;
  const float* yb = y + base;

  if (tid < 11) {
    // window[c][0][i][j] = g[i]*g[j]; recover g from row 5: g[k]=w[5][k]/sqrt(w[5][5])
    const float g5 = sqrtf(win[5 * 11 + 5]);
    gsh[tid] = win[5 * 11 + tid] / g5;
  }

  float* Hmu1 = ldsf;
  float* Hmu2 = ldsf + 1 * 28 * 256;
  float* Hxx  = ldsf + 2 * 28 * 256;
  float* Hyy  = ldsf + 3 * 28 * 256;
  float* Hxy  = ldsf + 4 * 28 * 256;

  const int lane = tid & 31;
  const int wid  = tid >> 5;
  const int i16  = lane & 15;
  const int hi   = lane >> 4;              // 0: K even-pair / M 0..7, 1: +2 / M 8..15

  const float C1 = 1.0e-4f;                // 0.01^2
  const float C2 = 9.0e-4f;                // 0.03^2
  float ssim_sum = 0.0f;

  for (int rt = 0; rt < 16; ++rt) {        // 16 row tiles of 16 output rows
    const int r0 = rt * 16;
    __syncthreads();                       // also publishes gsh on first iter

    // ---- horizontal pass: band rows t=0..27 (t>=26 are zero K-padding) ----
    for (int p = tid; p < 28 * 256; p += 256) {
      const int t   = p >> 8;
      const int col = p & 255;
      float h1 = 0.f, h2 = 0.f, hxx = 0.f, hyy = 0.f, hxy = 0.f;
      const int row = r0 + t - 5;          // zero padding outside image
      if (t < 26 && row >= 0 && row < 256) {
        const float* xr = xb + row * 256;
        const float* yr = yb + row * 256;
        #pragma unroll
        for (int k = 0; k < 11; ++k) {
          const int cc = col + k - 5;
          if (cc >= 0 && cc < 256) {
            const float xv = xr[cc], yv = yr[cc], gk = gsh[k];
            h1  += gk * xv;       h2  += gk * yv;
            hxx += gk * xv * xv;  hyy += gk * yv * yv;  hxy += gk * xv * yv;
          }
        }
      }
      Hmu1[p] = h1; Hmu2[p] = h2; Hxx[p] = hxx; Hyy[p] = hyy; Hxy[p] = hxy;
    }
    __syncthreads();

    // ---- vertical pass: per wave, two 16-col tiles; V = A(16x28) * H(28x16)
    for (int ct = 0; ct < 2; ++ct) {
      const int col0 = (wid * 2 + ct) * 16;
      v8f c_mu1 = {}, c_mu2 = {}, c_xx = {}, c_yy = {}, c_xy = {};
      #pragma unroll
      for (int k = 0; k < 7; ++k) {        // K=4 chunks, band rows 4k..4k+3
        const int tA = 4 * k + 2 * hi;
        // A[i][t] = g[t - i] (banded); lanes0-15: K=4k+{0,1}, lanes16-31: +{2,3}
        const int d0 = tA - i16;
        v2f a;
        a.x = ((unsigned)d0 <= 10u)       ? gsh[d0]     : 0.0f;
        a.y = ((unsigned)(d0 + 1) <= 10u) ? gsh[d0 + 1] : 0.0f;
        const int o = tA * 256 + col0 + i16;   // B: row K, col N=i16
        v2f bv;
        bv.x = Hmu1[o]; bv.y = Hmu1[o + 256];
        c_mu1 = __builtin_amdgcn_wmma_f32_16x16x4_f32(false, a, false, bv, (short)0, c_mu1, false, false);
        bv.x = Hmu2[o]; bv.y = Hmu2[o + 256];
        c_mu2 = __builtin_amdgcn_wmma_f32_16x16x4_f32(false, a, false, bv, (short)0, c_mu2, false, false);
        bv.x = Hxx[o];  bv.y = Hxx[o + 256];
        c_xx  = __builtin_amdgcn_wmma_f32_16x16x4_f32(false, a, false, bv, (short)0, c_xx,  false, false);
        bv.x = Hyy[o];  bv.y = Hyy[o + 256];
        c_yy  = __builtin_amdgcn_wmma_f32_16x16x4_f32(false, a, false, bv, (short)0, c_yy,  false, false);
        bv.x = Hxy[o];  bv.y = Hxy[o + 256];
        c_xy  = __builtin_amdgcn_wmma_f32_16x16x4_f32(false, a, false, bv, (short)0, c_xy,  false, false);
      }
      // D layout: element e -> pixel (row r0 + e + 8*hi, col col0 + i16)
      #pragma unroll
      for (int e = 0; e < 8; ++e) {
        const float m1 = c_mu1[e], m2 = c_mu2[e];
        const float m1s = m1 * m1, m2s = m2 * m2, m12 = m1 * m2;
        const float s1  = c_xx[e] - m1s;
        const float s2  = c_yy[e] - m2s;
        const float s12 = c_xy[e] - m12;
        ssim_sum += ((2.0f * m12 + C1) * (2.0f * s12 + C2)) /
                    ((m1s + m2s + C1) * (s1 + s2 + C2));
      }
    }
  }

  const float tot = block_reduce_sum(ssim_sum, sred);
  if (tid == 0) ws_partial[img] = tot;     // deterministic per-(b,c) partial
}

// ---------------------------------------------------------------------------
// Kernel 3: combine per-channel SSIM partials into out[b*3+1]
// ---------------------------------------------------------------------------
__global__ void ssim_finalize_kernel(const float* __restrict__ ws_partial,
                                     float* __restrict__ out) {
  const int b = blockIdx.x * blockDim.x + threadIdx.x;
  if (b < 64) {
    const float s = ws_partial[b * 3 + 0] + ws_partial[b * 3 + 1] + ws_partial[b * 3 + 2];
    out[b * 3 + 1] = s * (1.0f / (float)NPB);
  }
}

// ---------------------------------------------------------------------------
extern "C" void kernel_launch(void* const* d_in, const int* in_sizes, int n_in,
                              void* d_out, int out_size, void* d_ws, size_t ws_size,
                              hipStream_t stream) {
  const float* x   = (const float*)d_in[0];   // [64,3,256,256]
  const float* y   = (const float*)d_in[1];   // [64,3,256,256]
  const float* win = (const float*)d_in[2];   // [3,1,11,11]
  float* out = (float*)d_out;                 // [64,3,1]
  float* wsp = (float*)d_ws;                  // 192 f32 partials

  // Pass 1: MI + PSNR (256 KB LDS joint histogram per workgroup)
  mi_psnr_kernel<<<64, 1024, NBINS * NBINS * sizeof(unsigned int), stream>>>(x, y, out);

  // Pass 2: SSIM (L2-resident re-read of x/y; 140 KB LDS conv bands)
  ssim_kernel<<<192, 256, 5 * 28 * 256 * sizeof(float), stream>>>(x, y, win, wsp);

  // Pass 3: tiny finalize
  ssim_finalize_kernel<<<1, 64, 0, stream>>>(wsp, out);
}